// HAttentionLayer_34033320853962
// MI455X (gfx1250) — compile-verified
//
#include <hip/hip_runtime.h>
#include <hip/hip_bf16.h>

#define BB 32
#define NN 1024
#define HH 768
#define AA 8
#define TK 20

// ---- per-batch workspace layout (floats) ----
#define OFF_AIND 0
#define OFF_BIND (NN)
#define OFF_DV   (2*NN)
#define OFF_DE   (3*NN)            // 4 floats
#define OFF_HYP  (3*NN + 4)        // 3*NN
#define OFF_WA   (6*NN + 4)        // 3*NN  softmax weights
#define OFF_WG   (9*NN + 4)        // 3*NN  Hyper*Dv weights
#define OFF_HP   (12*NN + 4)       // 3*HH  Hyper_prime
#define OFF_G    (12*NN + 4 + 3*HH)
#define OFF_KH   (12*NN + 4 + 6*HH)
#define OFF_SE   (12*NN + 4 + 9*HH) // 4
#define WS_PER_B (12*NN + 8 + 9*HH)

typedef __attribute__((ext_vector_type(2))) float v2f;
typedef __attribute__((ext_vector_type(8))) float v8f;

// ------------------------------------------------------------------
// K1: indicators via per-wave top-20 of gathered att rows, then Dv/De
// grid: B blocks of 512 (16 waves; wave w -> one gathered row)
// ------------------------------------------------------------------
__global__ void k1_indicators(const float* __restrict__ att,
                              const int* __restrict__ a_idx,
                              const int* __restrict__ b_idx,
                              const float* __restrict__ speak_all,
                              float* __restrict__ ws) {
  const int b = blockIdx.x;
  float* w    = ws + (size_t)b * WS_PER_B;
  float* aind = w + OFF_AIND;
  float* bind = w + OFF_BIND;
  float* Dv   = w + OFF_DV;
  float* De   = w + OFF_DE;
  const int t = threadIdx.x;

  for (int i = t; i < 2 * NN; i += 512) w[i] = 0.0f;  // aind & bind contiguous
  __shared__ float rsum[3];
  if (t < 3) rsum[t] = 0.0f;
  __syncthreads();

  const int wv = t >> 5;
  const int lane = t & 31;
  const int src = (wv < 8) ? a_idx[b * AA + wv] : b_idx[b * AA + (wv - 8)];
  float* ind = (wv < 8) ? aind : bind;
  const float* row = att + (size_t)b * NN * NN + (size_t)src * NN;

  float vals[NN / 32];
#pragma unroll
  for (int c = 0; c < NN / 32; ++c) vals[c] = row[c * 32 + lane];
  const bool mask = (src != 0);

  for (int k = 0; k < TK; ++k) {
    float bv = -1.0f;
    int bj = NN;
#pragma unroll
    for (int c = 0; c < NN / 32; ++c) {
      int j = c * 32 + lane;
      float v = vals[c];
      if (v > bv || (v == bv && j < bj)) { bv = v; bj = j; }
    }
    for (int m = 16; m >= 1; m >>= 1) {
      float ov = __shfl_xor(bv, m, 32);
      int oj = __shfl_xor(bj, m, 32);
      if (ov > bv || (ov == bv && oj < bj)) { bv = ov; bj = oj; }
    }
    // bj uniform now; knock out the winning element (static reg indices)
    const int cc = bj >> 5;
    const bool owner = (lane == (bj & 31));
#pragma unroll
    for (int c = 0; c < NN / 32; ++c)
      vals[c] = (owner && c == cc) ? -1.0f : vals[c];
    if (lane == 0 && mask) ind[bj] = 1.0f;
  }
  __syncthreads();

  const float* spk = speak_all + (size_t)b * 3 * NN;  // row 0
  float p0 = 0.f, p1 = 0.f, p2 = 0.f;
  for (int n = t; n < NN; n += 512) {
    float av = aind[n], bv2 = bind[n], sv = spk[n];
    Dv[n] = rsqrtf(av + bv2 + sv + 1.0f);
    p0 += av; p1 += bv2; p2 += sv;
  }
  for (int m = 16; m >= 1; m >>= 1) {
    p0 += __shfl_xor(p0, m, 32);
    p1 += __shfl_xor(p1, m, 32);
    p2 += __shfl_xor(p2, m, 32);
  }
  if (lane == 0) {
    atomicAdd(&rsum[0], p0);
    atomicAdd(&rsum[1], p1);
    atomicAdd(&rsum[2], p2);
  }
  __syncthreads();
  if (t < 3) De[t] = rsqrtf(rsum[t]);
}

// ------------------------------------------------------------------
// K3: Hyper = Hyper0 @ att via V_WMMA_F32_16X16X4_F32 (M padded 3->16)
// grid: (8 col-groups, B), block 256 = 8 waves; wave -> one 16-col tile
// ------------------------------------------------------------------
__global__ void k3_hyper_wmma(const float* __restrict__ att,
                              const float* __restrict__ speak_all,
                              float* __restrict__ ws) {
  const int b = blockIdx.y;
  float* w = ws + (size_t)b * WS_PER_B;
  const float* aind = w + OFF_AIND;
  const float* bind = w + OFF_BIND;
  float* Hyp = w + OFF_HYP;

  __shared__ float ldsH[3][NN + 4];
  const int t = threadIdx.x;
  for (int i = t; i < NN; i += 256) {
    ldsH[0][i] = aind[i];
    ldsH[1][i] = bind[i];
    ldsH[2][i] = speak_all[(size_t)b * 3 * NN + i];
  }
  __syncthreads();

  const int wv = t >> 5, lane = t & 31;
  const int jc = (blockIdx.x * 8 + wv) * 16;
  const int row = lane & 15;          // M index (A) / N index (B)
  const int k0 = (lane >> 4) * 2;     // K offset for this half-wave
  const int col = jc + row;
  const float amask = (row < 3) ? 1.0f : 0.0f;
  const int r3 = (row < 3) ? row : 0;
  const float* attb = att + (size_t)b * NN * NN;

  v8f c = {};
  for (int i0 = 0; i0 < NN; i0 += 4) {
    v2f a, bf;
    a.x = amask * ldsH[r3][i0 + k0];
    a.y = amask * ldsH[r3][i0 + k0 + 1];
    bf.x = attb[(size_t)(i0 + k0) * NN + col];
    bf.y = attb[(size_t)(i0 + k0 + 1) * NN + col];
    c = __builtin_amdgcn_wmma_f32_16x16x4_f32(false, a, false, bf,
                                              (short)0, c, false, false);
  }
  if (lane < 16) {  // rows M=0..2 live in c[0..2] lanes 0-15
    Hyp[0 * NN + jc + lane] = c[0];
    Hyp[1 * NN + jc + lane] = c[1];
    Hyp[2 * NN + jc + lane] = c[2];
  }
}

// ------------------------------------------------------------------
// K4a: per-edge softmax weights Wa = softmax(Hyper), Wg = Hyper*Dv
// grid: B blocks of 128 (waves 0-2 active, one per edge)
// ------------------------------------------------------------------
__global__ void k4a_weights(float* __restrict__ ws) {
  const int b = blockIdx.x;
  float* w = ws + (size_t)b * WS_PER_B;
  const int t = threadIdx.x;
  const int wv = t >> 5, lane = t & 31;
  if (wv >= 3) return;
  const float* Hyp = w + OFF_HYP + wv * NN;
  float* Wa = w + OFF_WA + wv * NN;
  float* Wg = w + OFF_WG + wv * NN;
  const float* Dv = w + OFF_DV;

  float vals[NN / 32];
  float m = -3.0e38f;
#pragma unroll
  for (int c = 0; c < NN / 32; ++c) {
    vals[c] = Hyp[c * 32 + lane];
    m = fmaxf(m, vals[c]);
  }
  for (int s = 16; s >= 1; s >>= 1) m = fmaxf(m, __shfl_xor(m, s, 32));
  float sum = 0.f;
#pragma unroll
  for (int c = 0; c < NN / 32; ++c) {
    vals[c] = __expf(vals[c] - m);
    sum += vals[c];
  }
  for (int s = 16; s >= 1; s >>= 1) sum += __shfl_xor(sum, s, 32);
  const float inv = 1.0f / sum;
#pragma unroll
  for (int c = 0; c < NN / 32; ++c) {
    int j = c * 32 + lane;
    Wa[j] = vals[c] * inv;
    Wg[j] = Hyp[j] * Dv[j];
  }
}

// ------------------------------------------------------------------
// K4b: single pass over h: Hp[e] = sum_j Wa[e][j]*h[j], G[e] = sum_j Wg*h
// grid: (HH/256, B), block 256 (thread -> one h column)
// ------------------------------------------------------------------
__global__ void k4b_hsums(const float* __restrict__ h, float* __restrict__ ws) {
  const int b = blockIdx.y;
  float* w = ws + (size_t)b * WS_PER_B;
  __shared__ float lwa[3][NN];
  __shared__ float lwg[3][NN];
  const int t = threadIdx.x;
  const float* Wa = w + OFF_WA;
  const float* Wg = w + OFF_WG;
  for (int i = t; i < 3 * NN; i += 256) {
    (&lwa[0][0])[i] = Wa[i];
    (&lwg[0][0])[i] = Wg[i];
  }
  __syncthreads();

  const int hcol = blockIdx.x * 256 + t;
  const float* hb = h + (size_t)b * NN * HH + hcol;
  float hp0 = 0, hp1 = 0, hp2 = 0, g0 = 0, g1 = 0, g2 = 0;
  for (int j = 0; j < NN; ++j) {
    float v = hb[(size_t)j * HH];
    hp0 += lwa[0][j] * v; hp1 += lwa[1][j] * v; hp2 += lwa[2][j] * v;
    g0  += lwg[0][j] * v; g1  += lwg[1][j] * v; g2  += lwg[2][j] * v;
  }
  float* Hp = w + OFF_HP;
  float* G  = w + OFF_G;
  Hp[0 * HH + hcol] = hp0; Hp[1 * HH + hcol] = hp1; Hp[2 * HH + hcol] = hp2;
  G[0 * HH + hcol]  = g0;  G[1 * HH + hcol]  = g1;  G[2 * HH + hcol]  = g2;
}

// ------------------------------------------------------------------
// K5: edge softmax w, KH[e] = w[e]*De[e]*G[e], s_e = KH[e].S_w
// grid: B blocks of 256
// ------------------------------------------------------------------
__global__ void k5_edge(const float* __restrict__ W_w, const float* __restrict__ W_b,
                        const float* __restrict__ S_w, float* __restrict__ ws) {
  const int b = blockIdx.x;
  float* w = ws + (size_t)b * WS_PER_B;
  float* Hp = w + OFF_HP;
  float* G  = w + OFF_G;
  float* KH = w + OFF_KH;
  float* sE = w + OFF_SE;
  const float* De = w + OFF_DE;
  const int t = threadIdx.x, lane = t & 31, wv = t >> 5;

  __shared__ float rw[8][3];
  __shared__ float kfac[3];

  float p0 = 0, p1 = 0, p2 = 0;
  for (int i = t; i < HH; i += 256) {
    float wwv = W_w[i];
    p0 += Hp[i] * wwv;
    p1 += Hp[HH + i] * wwv;
    p2 += Hp[2 * HH + i] * wwv;
  }
  for (int s = 16; s >= 1; s >>= 1) {
    p0 += __shfl_xor(p0, s, 32);
    p1 += __shfl_xor(p1, s, 32);
    p2 += __shfl_xor(p2, s, 32);
  }
  if (lane == 0) { rw[wv][0] = p0; rw[wv][1] = p1; rw[wv][2] = p2; }
  __syncthreads();
  if (t == 0) {
    float t0 = 0, t1 = 0, t2 = 0;
    for (int i = 0; i < 8; ++i) { t0 += rw[i][0]; t1 += rw[i][1]; t2 += rw[i][2]; }
    float wb = W_b[0];
    t0 += wb; t1 += wb; t2 += wb;
    float mm = fmaxf(t0, fmaxf(t1, t2));
    float e0 = __expf(t0 - mm), e1 = __expf(t1 - mm), e2 = __expf(t2 - mm);
    float inv = 1.0f / (e0 + e1 + e2);
    kfac[0] = e0 * inv * De[0];
    kfac[1] = e1 * inv * De[1];
    kfac[2] = e2 * inv * De[2];
  }
  __syncthreads();
  const float k0 = kfac[0], k1 = kfac[1], k2 = kfac[2];
  float s0 = 0, s1 = 0, s2 = 0;
  for (int i = t; i < HH; i += 256) {
    float sw = S_w[i];
    float v0 = k0 * G[i], v1 = k1 * G[HH + i], v2 = k2 * G[2 * HH + i];
    KH[i] = v0; KH[HH + i] = v1; KH[2 * HH + i] = v2;
    s0 += v0 * sw; s1 += v1 * sw; s2 += v2 * sw;
  }
  for (int s = 16; s >= 1; s >>= 1) {
    s0 += __shfl_xor(s0, s, 32);
    s1 += __shfl_xor(s1, s, 32);
    s2 += __shfl_xor(s2, s, 32);
  }
  __syncthreads();
  if (lane == 0) { rw[wv][0] = s0; rw[wv][1] = s1; rw[wv][2] = s2; }
  __syncthreads();
  if (t == 0) {
    float a0 = 0, a1 = 0, a2 = 0;
    for (int i = 0; i < 8; ++i) { a0 += rw[i][0]; a1 += rw[i][1]; a2 += rw[i][2]; }
    sE[0] = a0; sE[1] = a1; sE[2] = a2;
  }
}

// ------------------------------------------------------------------
// K6: score[n] = Dv[n]*(aind*s0+bind*s1+spk*s2)+S_b; exact top-2;
//     emit X[top0], X[top1] analytically.  grid: B blocks of 512
// ------------------------------------------------------------------
__global__ void k6_score_top2(const float* __restrict__ speak_all,
                              const float* __restrict__ S_b,
                              const float* __restrict__ ws_c,
                              float* __restrict__ out) {
  const int b = blockIdx.x;
  const float* w = ws_c + (size_t)b * WS_PER_B;
  const float* aind = w + OFF_AIND;
  const float* bind = w + OFF_BIND;
  const float* Dv = w + OFF_DV;
  const float* KH = w + OFF_KH;
  const float* sE = w + OFF_SE;
  const float* spk = speak_all + (size_t)b * 3 * NN;
  const int t = threadIdx.x;

  __shared__ float sval[512];
  __shared__ int sidx[512];
  __shared__ int tops[2];

  const float sb = S_b[0];
  const float s0 = sE[0], s1 = sE[1], s2 = sE[2];

  for (int pass = 0; pass < 2; ++pass) {
    const int excl = (pass == 0) ? -1 : tops[0];
    float bv = -3.0e38f;
    int bj = NN;
    for (int n = t; n < NN; n += 512) {
      if (n == excl) continue;
      float sc = Dv[n] * (aind[n] * s0 + bind[n] * s1 + spk[n] * s2) + sb;
      if (sc > bv || (sc == bv && n < bj)) { bv = sc; bj = n; }
    }
    sval[t] = bv; sidx[t] = bj;
    __syncthreads();
    for (int s = 256; s >= 1; s >>= 1) {
      if (t < s) {
        float ov = sval[t + s]; int oj = sidx[t + s];
        if (ov > sval[t] || (ov == sval[t] && oj < sidx[t])) {
          sval[t] = ov; sidx[t] = oj;
        }
      }
      __syncthreads();
    }
    if (t == 0) tops[pass] = sidx[0];
    __syncthreads();
  }

  const int n0 = tops[0], n1 = tops[1];
  const float d0 = Dv[n0], a0 = aind[n0], b0 = bind[n0], c0 = spk[n0];
  const float d1 = Dv[n1], a1 = aind[n1], b1 = bind[n1], c1 = spk[n1];
  for (int hh = t; hh < HH; hh += 512) {
    float k0 = KH[hh], k1 = KH[HH + hh], k2 = KH[2 * HH + hh];
    out[(size_t)b * 2 * HH + hh]      = d0 * (a0 * k0 + b0 * k1 + c0 * k2);
    out[(size_t)b * 2 * HH + HH + hh] = d1 * (a1 * k0 + b1 * k1 + c1 * k2);
  }
}

extern "C" void kernel_launch(void* const* d_in, const int* in_sizes, int n_in,
                              void* d_out, int out_size, void* d_ws, size_t ws_size,
                              hipStream_t stream) {
  const float* inp   = (const float*)d_in[0];  // (B,N,H)
  const float* att   = (const float*)d_in[1];  // (B,1,N,N) == (B,N,N)
  const int*   a_idx = (const int*)d_in[2];    // (B,A)
  const int*   b_idx = (const int*)d_in[3];    // (B,A)
  // d_in[4] speak_num: unused by the reference computation
  const float* speak = (const float*)d_in[5];  // (B,3,N); row 0 used
  const float* W_w   = (const float*)d_in[6];  // (1,H)
  const float* W_b   = (const float*)d_in[7];  // (1,)
  const float* S_w   = (const float*)d_in[8];  // (1,H)
  const float* S_b   = (const float*)d_in[9];  // (1,)
  float* out = (float*)d_out;
  float* ws  = (float*)d_ws;

  k1_indicators<<<BB, 512, 0, stream>>>(att, a_idx, b_idx, speak, ws);
  k3_hyper_wmma<<<dim3(8, BB), 256, 0, stream>>>(att, speak, ws);
  k4a_weights<<<BB, 128, 0, stream>>>(ws);
  k4b_hsums<<<dim3(HH / 256, BB), 256, 0, stream>>>(inp, ws);
  k5_edge<<<BB, 256, 0, stream>>>(W_w, W_b, S_w, ws);
  k6_score_top2<<<BB, 512, 0, stream>>>(speak, S_b, ws, out);
}